// TfLSTM_3719441678700
// MI455X (gfx1250) — compile-verified
//
#include <hip/hip_runtime.h>
#include <math.h>

typedef __attribute__((ext_vector_type(2))) float v2f;
typedef __attribute__((ext_vector_type(8))) float v8f;

// ---------------------------------------------------------------------------
// Swizzled B-matrix layout for V_WMMA_F32_16X16X4_F32:
//   Bswz[((kb*NT + nt)*32 + lane)*2 + comp] = W[4*kb + (lane>>4)*2 + comp][nt*16 + (lane&15)]
// so each lane loads its 2-element B fragment with ONE coalesced b64 load.
// ---------------------------------------------------------------------------
__device__ inline void swz_decode(int e, int NT, int& k, int& col) {
  const int comp = e & 1;
  const int lane = (e >> 1) & 31;
  const int rest = e >> 6;
  const int nt = rest % NT;
  const int kb = rest / NT;
  k = 4 * kb + ((lane >> 4) << 1) + comp;
  col = nt * 16 + (lane & 15);
}

// Layer-1 weights: permute gate blocks [i|f|g|o] from 100 to 112 units
// (zero padding provably gives h=c=0 for padded units) AND swizzle.
__global__ void prep_layer1_kernel(const float* __restrict__ W1,
                                   const float* __restrict__ b1,
                                   const float* __restrict__ U1,
                                   float* __restrict__ W1s,   // [32*28*64]
                                   float* __restrict__ b1p,   // [448]
                                   float* __restrict__ U1s) { // [28*28*64]
  const int stride = gridDim.x * blockDim.x;
  const int idx = blockIdx.x * blockDim.x + threadIdx.x;
  for (int e = idx; e < 32 * 28 * 64; e += stride) {  // K=128, NT=28
    int k, col; swz_decode(e, 28, k, col);
    const int g = col / 112, u = col % 112;
    W1s[e] = (u < 100) ? W1[k * 400 + g * 100 + u] : 0.0f;
  }
  for (int e = idx; e < 448; e += stride) {
    const int g = e / 112, u = e % 112;
    b1p[e] = (u < 100) ? b1[g * 100 + u] : 0.0f;
  }
  for (int e = idx; e < 28 * 28 * 64; e += stride) {  // K=112, NT=28
    int k, col; swz_decode(e, 28, k, col);
    const int g = col / 112, u = col % 112;
    U1s[e] = (k < 100 && u < 100) ? U1[k * 400 + g * 100 + u] : 0.0f;
  }
}

// Layer-2 weights: H2=128 is 16-aligned, no permute; pad W2 rows 100..111, swizzle.
__global__ void prep_layer2_kernel(const float* __restrict__ W2,
                                   const float* __restrict__ U2,
                                   float* __restrict__ W2s,   // [28*32*64]
                                   float* __restrict__ U2s) { // [32*32*64]
  const int stride = gridDim.x * blockDim.x;
  const int idx = blockIdx.x * blockDim.x + threadIdx.x;
  for (int e = idx; e < 28 * 32 * 64; e += stride) {  // K=112, NT=32
    int k, col; swz_decode(e, 32, k, col);
    W2s[e] = (k < 100) ? W2[k * 512 + col] : 0.0f;
  }
  for (int e = idx; e < 32 * 32 * 64; e += stride) {  // K=128, NT=32
    int k, col; swz_decode(e, 32, k, col);
    U2s[e] = U2[k * 512 + col];
  }
}

// ---------------------------------------------------------------------------
// GEMM + bias, fp32 WMMA 16x16x4, output in WMMA C/D *fragment* layout:
//   Yfrag[((blk*NT + nt)*32 + lane)*8 .. +7]  (32B contiguous per lane)
// M-tiles are keyed (t, batch-tile): blk = t*32 + btile, rows = 16 batch rows
// at fixed t. A-row addressing via generic strides so both GEMMs share code.
// ---------------------------------------------------------------------------
__global__ __launch_bounds__(256)
void gemm_bias_frag_kernel(const float* __restrict__ X,
                           const float* __restrict__ Wswz,
                           const float* __restrict__ bias,
                           float* __restrict__ Yfrag,
                           int K, int NT, long rowStrideB, long rowStrideT) {
  const int NBT = 32;  // 512/16 batch tiles
  const int blk = blockIdx.x;
  const int t = blk / NBT, btile = blk % NBT;
  const int b0 = btile * 16;
  const int lane = threadIdx.x & 31;
  const int wave = threadIdx.x >> 5;
  const int lrow = lane & 15;
  const int khalf = (lane >> 4) * 2;

  const float* arow =
      X + (size_t)(b0 + lrow) * rowStrideB + (size_t)t * rowStrideT;
  const int KB = K >> 2;

  for (int nt = wave; nt < NT; nt += 8) {
    const float bv = bias[nt * 16 + lrow];
    v8f acc;
#pragma unroll
    for (int r = 0; r < 8; ++r) acc[r] = bv;

    for (int kb = 0; kb < KB; ++kb) {
      v2f a;
      a[0] = arow[4 * kb + khalf];
      a[1] = arow[4 * kb + khalf + 1];
      const v2f b =
          *(const v2f*)(Wswz + (((size_t)kb * NT + nt) * 32 + lane) * 2);
      acc = __builtin_amdgcn_wmma_f32_16x16x4_f32(false, a, false, b,
                                                  (short)0, acc, false, false);
    }
    *(v8f*)(Yfrag + (((size_t)blk * NT + nt) * 32 + lane) * 8) = acc;
  }
}

// ---------------------------------------------------------------------------
// Persistent LSTM scan. Wave w owns unit block w; its 4 gate z-tiles live in
// registers. zx arrives in fragment layout and is prefetched one timestep
// ahead into a per-wave LDS slab with GLOBAL_LOAD_ASYNC_TO_LDS_B128
// (ASYNCcnt), overlapping the h@U WMMA K-loop. h lives in LDS.
// ---------------------------------------------------------------------------
template <int H, bool STORE_ALL>
__global__ __launch_bounds__(256)
void lstm_scan_kernel(const float* __restrict__ ZXf,  // frag [T][32][NT][32][8]
                      const float* __restrict__ Uswz, // [H/4][NT][32][2]
                      float* __restrict__ Hout, int T) {
  constexpr int NB = H / 16;
  constexpr int NT = 4 * NB;
  constexpr int HP = H + 4;
  constexpr int NBT = 32;
  __shared__ float h_lds[16][HP];
  __shared__ float zslab[8 * 4 * 256];  // 4KB per wave

  const int btile = blockIdx.x;
  const int b0 = btile * 16;
  const int tid = threadIdx.x;
  const int lane = tid & 31;
  const int wave = tid >> 5;
  const int lrow = lane & 15;
  const int khalf = (lane >> 4) * 2;
  const int rbase = (lane < 16) ? 0 : 8;
  const bool active = (wave < NB);
  const int u0 = wave * 16;

  float* const myslab = &zslab[(wave * 4) * 256 + lane * 8];

  for (int e = tid; e < 16 * HP; e += 256) (&h_lds[0][0])[e] = 0.0f;

  v8f c_frag;
#pragma unroll
  for (int r = 0; r < 8; ++r) c_frag[r] = 0.0f;

  // issue async prefetch of zx fragments for t=0 (per-wave private slab)
  if (active) {
#pragma unroll
    for (int g = 0; g < 4; ++g) {
      const unsigned long long src = (unsigned long long)(
          ZXf + (((size_t)0 * NBT + btile) * NT + (g * NB + wave)) * 256 +
          lane * 8);
      const unsigned dst = (unsigned)(uintptr_t)(myslab + g * 256);
      asm volatile(
          "global_load_async_to_lds_b128 %0, %1, off\n\t"
          "global_load_async_to_lds_b128 %0, %1, off offset:16"
          :: "v"(dst), "v"(src) : "memory");
    }
  }
  __syncthreads();

  for (int t = 0; t < T; ++t) {
    v8f acc[4];
    if (active) {
      // wait for this wave's async zx slab, pull into accumulators
      asm volatile("s_wait_asynccnt 0x0" ::: "memory");
#pragma unroll
      for (int g = 0; g < 4; ++g)
        acc[g] = *(const v8f*)(myslab + g * 256);
      // make sure LDS reads landed in VGPRs before the slab is overwritten
      asm volatile("s_wait_dscnt 0x0" ::: "memory");
      // prefetch next timestep's zx while we run the recurrence matmul
      if (t + 1 < T) {
#pragma unroll
        for (int g = 0; g < 4; ++g) {
          const unsigned long long src = (unsigned long long)(
              ZXf +
              (((size_t)(t + 1) * NBT + btile) * NT + (g * NB + wave)) * 256 +
              lane * 8);
          const unsigned dst = (unsigned)(uintptr_t)(myslab + g * 256);
          asm volatile(
              "global_load_async_to_lds_b128 %0, %1, off\n\t"
              "global_load_async_to_lds_b128 %0, %1, off offset:16"
              :: "v"(dst), "v"(src) : "memory");
        }
      }
      // z += h @ U  (fp32 WMMA, swizzled b64 B-fragments)
      for (int kb = 0; kb < H / 4; ++kb) {
        v2f a;
        a[0] = h_lds[lrow][4 * kb + khalf];
        a[1] = h_lds[lrow][4 * kb + khalf + 1];
#pragma unroll
        for (int g = 0; g < 4; ++g) {
          const int nt = g * NB + wave;
          const v2f b =
              *(const v2f*)(Uswz + (((size_t)kb * NT + nt) * 32 + lane) * 2);
          acc[g] = __builtin_amdgcn_wmma_f32_16x16x4_f32(
              false, a, false, b, (short)0, acc[g], false, false);
        }
      }
    }
    __syncthreads();  // all waves done reading h_lds for this step

    if (active) {
      // gate / cell update entirely in registers
#pragma unroll
      for (int r = 0; r < 8; ++r) {
        const float zi = acc[0][r];
        const float zf = acc[1][r];
        const float zg = acc[2][r];
        const float zo = acc[3][r];
        const float ig = 1.0f / (1.0f + __expf(-zi));
        const float fg = 1.0f / (1.0f + __expf(-zf));
        const float gg = fmaxf(zg, 0.0f);  // activation='relu'
        const float og = 1.0f / (1.0f + __expf(-zo));
        const float c = fg * c_frag[r] + ig * gg;
        c_frag[r] = c;
        const float h = og * fmaxf(c, 0.0f);
        const int b = rbase + r;
        const int u = u0 + lrow;
        h_lds[b][u] = h;
        if (STORE_ALL) {
          // layout [t][b][H] so GEMM-2 A-rows are contiguous
          Hout[((size_t)t * 512 + (b0 + b)) * H + u] = h;
        } else if (t == T - 1) {
          Hout[(size_t)(b0 + b) * H + u] = h;
        }
      }
    }
    __syncthreads();  // h_lds valid for next step
  }
}

// ---------------------------------------------------------------------------
// Dense [512,128]@[128,19] + bias + softmax. One thread per batch row.
// ---------------------------------------------------------------------------
__global__ __launch_bounds__(256)
void dense_softmax_kernel(const float* __restrict__ H2,
                          const float* __restrict__ Wd,
                          const float* __restrict__ bd,
                          float* __restrict__ Out) {
  const int b = blockIdx.x * blockDim.x + threadIdx.x;
  if (b >= 512) return;
  float logits[19];
#pragma unroll
  for (int c = 0; c < 19; ++c) logits[c] = bd[c];
  for (int k = 0; k < 128; ++k) {
    const float hv = H2[b * 128 + k];
#pragma unroll
    for (int c = 0; c < 19; ++c) logits[c] += hv * Wd[k * 19 + c];
  }
  float mx = logits[0];
#pragma unroll
  for (int c = 1; c < 19; ++c) mx = fmaxf(mx, logits[c]);
  float s = 0.0f;
#pragma unroll
  for (int c = 0; c < 19; ++c) {
    logits[c] = __expf(logits[c] - mx);
    s += logits[c];
  }
  const float inv = 1.0f / s;
#pragma unroll
  for (int c = 0; c < 19; ++c) Out[b * 19 + c] = logits[c] * inv;
}

// ---------------------------------------------------------------------------
extern "C" void kernel_launch(void* const* d_in, const int* in_sizes, int n_in,
                              void* d_out, int out_size, void* d_ws,
                              size_t ws_size, hipStream_t stream) {
  const float* x  = (const float*)d_in[0];
  const float* W1 = (const float*)d_in[1];
  const float* U1 = (const float*)d_in[2];
  const float* b1 = (const float*)d_in[3];
  const float* W2 = (const float*)d_in[4];
  const float* U2 = (const float*)d_in[5];
  const float* b2 = (const float*)d_in[6];
  const float* Wd = (const float*)d_in[7];
  const float* bd = (const float*)d_in[8];
  float* out = (float*)d_out;
  float* ws = (float*)d_ws;

  constexpr int B = 512, T = 256;
  constexpr int H1P = 112, H2u = 128;
  constexpr size_t M = (size_t)B * T;

  // Workspace (floats); zx fragment region reused by both layers.
  size_t off = 0;
  float* zx  = ws + off; off += M * 512;            // max(M*448, M*512)
  float* h1  = ws + off; off += M * H1P;            // [T][B][112]
  float* h2  = ws + off; off += (size_t)B * H2u;    // [512][128]
  float* W1s = ws + off; off += (size_t)32 * 28 * 64;
  float* b1p = ws + off; off += 448;
  float* U1s = ws + off; off += (size_t)28 * 28 * 64;
  float* W2s = ws + off; off += (size_t)28 * 32 * 64;
  float* U2s = ws + off; off += (size_t)32 * 32 * 64;
  (void)ws_size; (void)in_sizes; (void)n_in; (void)out_size;

  // 1) permute/pad/swizzle weights
  prep_layer1_kernel<<<dim3(64), dim3(256), 0, stream>>>(W1, b1, U1, W1s, b1p, U1s);
  prep_layer2_kernel<<<dim3(64), dim3(256), 0, stream>>>(W2, U2, W2s, U2s);

  // 2) zx1 = x @ W1p + b1p  (fragment layout, NT=28)
  //    x rows: (b*T + t)*128  -> rowStrideB = T*128, rowStrideT = 128
  gemm_bias_frag_kernel<<<dim3(T * 32), dim3(256), 0, stream>>>(
      x, W1s, b1p, zx, 128, 28, (long)T * 128, 128L);

  // 3) layer-1 recurrence (async zx prefetch, stores h1 as [t][b][112])
  lstm_scan_kernel<112, true><<<dim3(B / 16), dim3(256), 0, stream>>>(
      zx, U1s, h1, T);

  // 4) zx2 = h1 @ W2p + b2  (fragment layout, NT=32)
  //    h1 rows: (t*512 + b)*112 -> rowStrideB = 112, rowStrideT = 512*112
  gemm_bias_frag_kernel<<<dim3(T * 32), dim3(256), 0, stream>>>(
      h1, W2s, b2, zx, 112, 32, 112L, (long)512 * 112);

  // 5) layer-2 recurrence (last hidden only)
  lstm_scan_kernel<128, false><<<dim3(B / 16), dim3(256), 0, stream>>>(
      zx, U2s, h2, T);

  // 6) dense + softmax
  dense_softmax_kernel<<<dim3(2), dim3(256), 0, stream>>>(h2, Wd, bd, out);
}